// SingleUserSequenceModel_47931835023400
// MI455X (gfx1250) — compile-verified
//
#include <hip/hip_runtime.h>

// Problem constants (from the reference)
#define T_SEQ 4096
#define T_B   4096
#define T_IN  2
#define T_H   3

typedef __attribute__((ext_vector_type(2))) float v2f;
typedef __attribute__((ext_vector_type(8))) float v8f;

// gfx1250 hardware tanh (V_TANH_F32) — confirmed present in round-1 asm.
__device__ __forceinline__ float hw_tanh(float x) {
#if __has_builtin(__builtin_amdgcn_tanhf)
    return __builtin_amdgcn_tanhf(x);
#elif __has_builtin(__builtin_amdgcn_tanh_f32)
    return __builtin_amdgcn_tanh_f32(x);
#else
    float e = __builtin_amdgcn_exp2f(x * 2.885390081777927f); // 2*log2(e)
    return 1.0f - 2.0f * __builtin_amdgcn_rcpf(e + 1.0f);
#endif
}

__device__ __forceinline__ float fast_sigmoid(float z) {
    float e = __builtin_amdgcn_exp2f(-z * 1.4426950408889634f); // log2(e)
    return __builtin_amdgcn_rcpf(1.0f + e);
}

// One wave owns 32 batch chains as two independent 16-chain WMMA pipelines.
//
// Per step, per group (16 chains = 16 N-columns of a 16x16x4 f32 WMMA):
//   D  = A_ih(16x4) x B_x(4x16) + 0      ; x-part: independent, issued early
//   D  = A_hh(16x4) x B_h(4x16) + D      ; recurrent part: the serial chain
//   h_j = tanh(D row j)
// A_hh rows j: [Whh[j][0..2], b_ih[j]+b_hh[j]]  with B row K=3 == 1.0
// A_ih rows j: [Wih[j][0..1], 0, 0]
// A rows 8..10 duplicate rows 0..2, so D rows 8..10 (lanes 16-31) duplicate
// the h values -> next step's B operand is just two v_cndmask selects, no
// cross-lane ops anywhere in the dependent chain.
__global__ __launch_bounds__(32) void rnn_wmma_kernel(
    const float* __restrict__ x,
    const float* __restrict__ W_ih,
    const float* __restrict__ W_hh,
    const float* __restrict__ b_ih,
    const float* __restrict__ b_hh,
    const float* __restrict__ W_lin,
    const float* __restrict__ b_lin,
    float* __restrict__ out)
{
    const int  lane  = threadIdx.x & 31;
    const int  bbase = blockIdx.x * 32;   // 32 chains per wave
    const int  m     = lane & 15;         // A-matrix row held by this lane
    const bool hi    = lane >= 16;

    // Weight rows live in A rows {0,1,2} and duplicated {8,9,10}; others zero.
    const bool valid = (m < T_H) || (m >= 8 && m < 8 + T_H);
    const int  r     = valid ? (m & 7) : 0;

    const float whh0 = valid ? W_hh[r * T_H + 0] : 0.0f;
    const float whh1 = valid ? W_hh[r * T_H + 1] : 0.0f;
    const float whh2 = valid ? W_hh[r * T_H + 2] : 0.0f;
    const float bias = valid ? (b_ih[r] + b_hh[r]) : 0.0f;
    const float wih0 = valid ? W_ih[r * T_IN + 0] : 0.0f;
    const float wih1 = valid ? W_ih[r * T_IN + 1] : 0.0f;

    // f32 16x4 A layout: VGPR0 = {K0 lo-lanes, K2 hi-lanes}, VGPR1 = {K1, K3}
    v2f a_hh, a_ih;
    a_hh.x = hi ? whh2 : whh0;  // K2 : K0
    a_hh.y = hi ? bias : whh1;  // K3 : K1  (B row K3 == 1.0 adds bias)
    a_ih.x = hi ? 0.0f : wih0;  // K2 unused (A col = 0)
    a_ih.y = hi ? 0.0f : wih1;  // K3 unused

    // Recurrent B operand, h0 = 0:
    // B VGPR0 = {h0 lo-lanes, h2 hi-lanes}, B VGPR1 = {h1 lo, 1.0 hi}
    v2f bhA, bhB;
    bhA.x = 0.0f;
    bhA.y = hi ? 1.0f : 0.0f;
    bhB = bhA;

    const v8f czero = {};

    // x is [SEQ, BATCH, 2]. Both half-waves load BOTH groups' float2 from the
    // same base register (imm offsets 0 / +128B): duplicated addresses within
    // the wave coalesce, and no cross-lane shuffles are needed.
    const float* xp      = x + (size_t)(bbase + m) * T_IN;   // group A chain
    const size_t tstride = (size_t)T_B * T_IN;               // floats per step

    constexpr int PF = 8;   // prefetch depth: 8 steps = 256 KiB ahead
    v2f xbA[PF], xbB[PF];
#pragma unroll
    for (int i = 0; i < PF; ++i) {
        xbA[i] = __builtin_nontemporal_load((const v2f*)(xp + (size_t)i * tstride));
        xbB[i] = __builtin_nontemporal_load((const v2f*)(xp + (size_t)i * tstride + 32));
    }

    float tA0 = 0.f, tA1 = 0.f, tA2 = 0.f;
    float tB0 = 0.f, tB1 = 0.f, tB2 = 0.f;

#define RNN_STEP(xvA, xvB)                                                     \
    do {                                                                       \
        /* x-part first: independent of the recurrence, can issue early */     \
        v8f accA = __builtin_amdgcn_wmma_f32_16x16x4_f32(                      \
            false, a_ih, false, (xvA), (short)0, czero, false, false);         \
        v8f accB = __builtin_amdgcn_wmma_f32_16x16x4_f32(                      \
            false, a_ih, false, (xvB), (short)0, czero, false, false);         \
        /* recurrent part: the serial dependent chain */                       \
        accA = __builtin_amdgcn_wmma_f32_16x16x4_f32(                          \
            false, a_hh, false, bhA, (short)0, accA, false, false);            \
        tA0 = hw_tanh(accA[0]);                                                \
        tA1 = hw_tanh(accA[1]);                                                \
        tA2 = hw_tanh(accA[2]);                                                \
        bhA.x = hi ? tA2 : tA0;                                                \
        bhA.y = hi ? 1.0f : tA1;                                               \
        accB = __builtin_amdgcn_wmma_f32_16x16x4_f32(                          \
            false, a_hh, false, bhB, (short)0, accB, false, false);            \
        tB0 = hw_tanh(accB[0]);                                                \
        tB1 = hw_tanh(accB[1]);                                                \
        tB2 = hw_tanh(accB[2]);                                                \
        bhB.x = hi ? tB2 : tB0;                                                \
        bhB.y = hi ? 1.0f : tB1;                                               \
    } while (0)

    // Steady-state: unconditional prefetch PF ahead (no branches, no buffer
    // rotation). (T_SEQ - PF) = 4088 is a multiple of PF.
    for (int tb = 0; tb < T_SEQ - PF; tb += PF) {
#pragma unroll
        for (int u = 0; u < PF; ++u) {
            v2f xvA = xbA[u];
            v2f xvB = xbB[u];
            const float* pn = xp + (size_t)(tb + u + PF) * tstride;
            xbA[u] = __builtin_nontemporal_load((const v2f*)pn);
            xbB[u] = __builtin_nontemporal_load((const v2f*)(pn + 32));
            RNN_STEP(xvA, xvB);
        }
    }
    // Epilogue: last PF steps, no loads.
#pragma unroll
    for (int u = 0; u < PF; ++u)
        RNN_STEP(xbA[u], xbB[u]);
#undef RNN_STEP

    // Epilogue output: lanes 0-15 hold group A's h at N=lane; lanes 16-31 hold
    // group B's h at N=lane-16 (duplicated rows 8-10) -> out[bbase+lane].
    float h0 = hi ? tB0 : tA0;
    float h1 = hi ? tB1 : tA1;
    float h2 = hi ? tB2 : tA2;
    h0 = fmaxf(h0, 0.0f);
    h1 = fmaxf(h1, 0.0f);
    h2 = fmaxf(h2, 0.0f);
    float z = h0 * W_lin[0] + h1 * W_lin[1] + h2 * W_lin[2] + b_lin[0];
    out[bbase + lane] = fast_sigmoid(z);
}

extern "C" void kernel_launch(void* const* d_in, const int* in_sizes, int n_in,
                              void* d_out, int out_size, void* d_ws, size_t ws_size,
                              hipStream_t stream) {
    const float* x     = (const float*)d_in[0];
    const float* W_ih  = (const float*)d_in[1];
    const float* W_hh  = (const float*)d_in[2];
    const float* b_ih  = (const float*)d_in[3];
    const float* b_hh  = (const float*)d_in[4];
    const float* W_lin = (const float*)d_in[5];
    const float* b_lin = (const float*)d_in[6];
    float* out = (float*)d_out;

    // 4096 chains / 32 per wave = 128 single-wave workgroups -> spread over up
    // to 128 WGPs so every wave gets its own SIMD32 for the serial recurrence.
    dim3 grid(T_B / 32), block(32);
    hipLaunchKernelGGL(rnn_wmma_kernel, grid, block, 0, stream,
                       x, W_ih, W_hh, b_ih, b_hh, W_lin, b_lin, out);
}